// TaylorAttention_73100343378332
// MI455X (gfx1250) — compile-verified
//
#include <hip/hip_runtime.h>
#include <hip/hip_bf16.h>

typedef __attribute__((ext_vector_type(2))) float v2f;
typedef __attribute__((ext_vector_type(4))) float v4f;
typedef __attribute__((ext_vector_type(8))) float v8f;

// D = A(16x4) * B(4x16) + C, fp32 tensor op on gfx1250
__device__ __forceinline__ v8f wmma4(v2f a, v2f b, v8f c) {
  return __builtin_amdgcn_wmma_f32_16x16x4_f32(
      false, a, false, b, (short)0, c, false, false);
}

#define BATCH 2
#define SEQ   8192
#define CH    512
#define HEADS 8
#define HD    64
#define MTOT  (BATCH * SEQ)    // 16384
#define NTOT  (3 * CH)         // 1536
#define SCALE 0.125f           // 64^-0.5
#define NSCALE 1024.0f         // SEQ * SCALE
#define LP    20               // LDS row pitch (floats): 80B = 5*16B, conflict-free

// ---------------------------------------------------------------------------
// Kernel 1: qkv = x @ W^T, scattered directly into q/k/v as [B,H,N,d]
// Block tile 128x128, K-chunk 16. 256 threads = 8 waves (2 M-slots x 4 N-slots),
// each wave computes 64x32 via 4x2 WMMA 16x16 tiles. b128 staging.
// ---------------------------------------------------------------------------
__global__ __launch_bounds__(256) void qkv_gemm(
    const float* __restrict__ x, const float* __restrict__ W,
    float* __restrict__ q, float* __restrict__ k, float* __restrict__ v) {
  __shared__ __align__(16) float As[128][LP];
  __shared__ __align__(16) float Bs[128][LP];

  const int tid  = threadIdx.x;
  const int wid  = tid >> 5;
  const int lane = tid & 31;
  const int l16  = lane & 15;
  const int lhi  = lane >> 4;      // 0: K pair {0,1}, 1: K pair {2,3}
  const int wm   = wid & 1;        // M slot (64 rows)
  const int wn   = wid >> 1;       // N slot (32 cols)
  const int m_block = blockIdx.y * 128;
  const int c_block = blockIdx.x * 128;

  const v8f z = {0.f,0.f,0.f,0.f,0.f,0.f,0.f,0.f};
  v8f acc[4][2];
#pragma unroll
  for (int i = 0; i < 4; ++i)
#pragma unroll
    for (int j = 0; j < 2; ++j) acc[i][j] = z;

  for (int k0 = 0; k0 < CH; k0 += 16) {
    // 128x16 tile = 512 float4 per matrix; 256 threads -> 2 float4 each
#pragma unroll
    for (int i = 0; i < 2; ++i) {
      int idx = tid + i * 256;               // 0..511
      int r = idx >> 2;                      // row 0..127
      int c = (idx & 3) << 2;                // col 0,4,8,12
      *(v4f*)&As[r][c] = *(const v4f*)&x[(m_block + r) * CH + k0 + c];
      *(v4f*)&Bs[r][c] = *(const v4f*)&W[(c_block + r) * CH + k0 + c];
    }
    __syncthreads();
#pragma unroll
    for (int ks = 0; ks < 16; ks += 4) {
      const int kk = ks + lhi * 2;
      v2f af[4], bf[2];
#pragma unroll
      for (int i = 0; i < 4; ++i) {
        int row = wm * 64 + i * 16 + l16;
        af[i].x = As[row][kk];
        af[i].y = As[row][kk + 1];
      }
#pragma unroll
      for (int j = 0; j < 2; ++j) {
        int row = wn * 32 + j * 16 + l16;
        bf[j].x = Bs[row][kk];
        bf[j].y = Bs[row][kk + 1];
      }
#pragma unroll
      for (int i = 0; i < 4; ++i)
#pragma unroll
        for (int j = 0; j < 2; ++j)
          acc[i][j] = wmma4(af[i], bf[j], acc[i][j]);
    }
    __syncthreads();
  }

  // Scatter: column c -> (t = c/512, h = (c%512)/64, dd = c%64)
#pragma unroll
  for (int i = 0; i < 4; ++i)
#pragma unroll
    for (int j = 0; j < 2; ++j) {
      int cg  = c_block + wn * 32 + j * 16 + l16;
      int t   = cg >> 9;
      int rem = cg & 511;
      int h   = rem >> 6;
      int dd  = rem & 63;
      float* base = (t == 0) ? q : (t == 1) ? k : v;
#pragma unroll
      for (int r = 0; r < 8; ++r) {
        int mg = m_block + wm * 64 + i * 16 + lhi * 8 + r;
        int b  = mg >> 13;
        int n  = mg & (SEQ - 1);
        base[(((b << 3) + h) * SEQ + n) * HD + dd] = acc[i][j][r];
      }
    }
}

// ---------------------------------------------------------------------------
// Kernel 2: per (b,h): g = K^T V - mean(K) (x) sum(V)   [64x64], k_hat_sum [64]
// One block per (b,h), 512 threads = 16 waves; each wave does K^T V over a
// 512-row chunk with 4x4 WMMA tiles (K-dim = sequence), reduce via LDS atomics.
// ---------------------------------------------------------------------------
__global__ __launch_bounds__(512) void kv_agg(
    const float* __restrict__ kp, const float* __restrict__ vp,
    float* __restrict__ g, float* __restrict__ ks_out) {
  __shared__ float gS[64 * 64];
  __shared__ float ksS[64];
  __shared__ float vsS[64];

  const int bh  = blockIdx.x;
  const float* K = kp + (size_t)bh * SEQ * HD;
  const float* V = vp + (size_t)bh * SEQ * HD;
  const int tid = threadIdx.x;

  for (int i = tid; i < 64 * 64; i += 512) gS[i] = 0.0f;
  if (tid < 64) { ksS[tid] = 0.0f; vsS[tid] = 0.0f; }
  __syncthreads();

  // ---- column sums of K and V over the sequence (b128 loads)
  {
    int d4 = (tid & 15) << 2;                 // 0,4,...,60
    int r0 = tid >> 4;                        // 0..31
    v4f sk = {0.f,0.f,0.f,0.f}, sv = {0.f,0.f,0.f,0.f};
    for (int n = r0; n < SEQ; n += 32) {
      sk += *(const v4f*)&K[n * HD + d4];
      sv += *(const v4f*)&V[n * HD + d4];
    }
    atomicAdd(&ksS[d4 + 0], sk.x); atomicAdd(&ksS[d4 + 1], sk.y);
    atomicAdd(&ksS[d4 + 2], sk.z); atomicAdd(&ksS[d4 + 3], sk.w);
    atomicAdd(&vsS[d4 + 0], sv.x); atomicAdd(&vsS[d4 + 1], sv.y);
    atomicAdd(&vsS[d4 + 2], sv.z); atomicAdd(&vsS[d4 + 3], sv.w);
  }

  // ---- WMMA partial K^T V
  const int wid = tid >> 5, lane = tid & 31, l16 = lane & 15, lhi = lane >> 4;
  const v8f z = {0.f,0.f,0.f,0.f,0.f,0.f,0.f,0.f};
  v8f acc[4][4];
#pragma unroll
  for (int i = 0; i < 4; ++i)
#pragma unroll
    for (int j = 0; j < 4; ++j) acc[i][j] = z;

  const int nbase = wid * 512;
  for (int s = 0; s < 512; s += 4) {
    const int n0 = nbase + s;
    const float* Kr = K + (size_t)(n0 + lhi * 2) * HD;   // rows n0+{0,1} / {2,3}
    const float* Vr = V + (size_t)(n0 + lhi * 2) * HD;
    v2f af[4], bf[4];
#pragma unroll
    for (int i = 0; i < 4; ++i) {            // A[m][kk] = K[n0+kk][m]
      af[i].x = Kr[i * 16 + l16];
      af[i].y = Kr[HD + i * 16 + l16];
    }
#pragma unroll
    for (int j = 0; j < 4; ++j) {            // B[kk][e] = V[n0+kk][e]
      bf[j].x = Vr[j * 16 + l16];
      bf[j].y = Vr[HD + j * 16 + l16];
    }
#pragma unroll
    for (int i = 0; i < 4; ++i)
#pragma unroll
      for (int j = 0; j < 4; ++j)
        acc[i][j] = wmma4(af[i], bf[j], acc[i][j]);
  }

  // ---- cross-wave reduce into gS
#pragma unroll
  for (int i = 0; i < 4; ++i)
#pragma unroll
    for (int j = 0; j < 4; ++j) {
      int e = j * 16 + l16;
#pragma unroll
      for (int r = 0; r < 8; ++r) {
        int m = i * 16 + lhi * 8 + r;
        atomicAdd(&gS[m * 64 + e], acc[i][j][r]);
      }
    }
  __syncthreads();

  // ---- finalize: g = K^T V - mean (x) vsum ; k_hat_sum = ksum - N*mean
  float* Gout = g + bh * 64 * 64;
  const float invN = 1.0f / (float)SEQ;
  for (int i = tid; i < 64 * 64; i += 512) {
    int d = i >> 6, e = i & 63;
    Gout[i] = gS[i] - (ksS[d] * invN) * vsS[e];
  }
  if (tid < 64) {
    float mean = ksS[tid] * invN;
    ks_out[bh * 64 + tid] = ksS[tid] - (float)SEQ * mean;
  }
}

// ---------------------------------------------------------------------------
// Kernel 3: z = (scale + Q @ g) / (N*scale + Q . k_hat_sum)
// Block = (b,h) x 128-row chunk, 256 threads = 8 waves, each wave a 16-row
// WMMA tile against g held in LDS.
// ---------------------------------------------------------------------------
__global__ __launch_bounds__(256) void out_kernel(
    const float* __restrict__ qp, const float* __restrict__ g,
    const float* __restrict__ ks, float* __restrict__ out) {
  __shared__ __align__(16) float gS[64 * 64];
  __shared__ float ksS[64];
  __shared__ float tdS[128];

  const int bh    = blockIdx.x;
  const int chunk = blockIdx.y;
  const float* Q  = qp + (size_t)bh * SEQ * HD;
  const float* G  = g + bh * 64 * 64;
  const int tid   = threadIdx.x;

#pragma unroll
  for (int i = 0; i < 4; ++i)                // 4096 floats = 1024 float4
    ((v4f*)gS)[tid + i * 256] = ((const v4f*)G)[tid + i * 256];
  if (tid < 64) ksS[tid] = ks[bh * 64 + tid];
  __syncthreads();

  if (tid < 128) {                           // t_D per row
    int m = chunk * 128 + tid;
    const float* Qr = Q + (size_t)m * HD;
    float s = 0.f;
#pragma unroll
    for (int d = 0; d < HD / 4; ++d) {
      v4f qv = *(const v4f*)&Qr[d * 4];
      s += qv.x * ksS[d * 4 + 0] + qv.y * ksS[d * 4 + 1] +
           qv.z * ksS[d * 4 + 2] + qv.w * ksS[d * 4 + 3];
    }
    tdS[tid] = NSCALE + s;
  }
  __syncthreads();

  const int wid = tid >> 5, lane = tid & 31, l16 = lane & 15, lhi = lane >> 4;
  const int m0  = chunk * 128 + wid * 16;
  const v8f z = {0.f,0.f,0.f,0.f,0.f,0.f,0.f,0.f};
  v8f acc[4];
#pragma unroll
  for (int j = 0; j < 4; ++j) acc[j] = z;

#pragma unroll
  for (int d0 = 0; d0 < HD; d0 += 4) {
    // A[mm][kk] = Q[m0+mm][d0+kk]; per-lane pair is contiguous -> 8B load
    v2f af = *(const v2f*)(Q + (size_t)(m0 + l16) * HD + d0 + lhi * 2);
    const int dd = d0 + lhi * 2;
#pragma unroll
    for (int j = 0; j < 4; ++j) {            // B[kk][e] = g[dd][e]
      v2f bf;
      bf.x = gS[dd * 64 + j * 16 + l16];
      bf.y = gS[(dd + 1) * 64 + j * 16 + l16];
      acc[j] = wmma4(af, bf, acc[j]);
    }
  }

#pragma unroll
  for (int j = 0; j < 4; ++j) {
    int e = j * 16 + l16;
#pragma unroll
    for (int r = 0; r < 8; ++r) {
      int mloc = wid * 16 + lhi * 8 + r;
      int m    = chunk * 128 + mloc;
      out[((size_t)bh * SEQ + m) * HD + e] = (SCALE + acc[j][r]) / tdS[mloc];
    }
  }
}

extern "C" void kernel_launch(void* const* d_in, const int* in_sizes, int n_in,
                              void* d_out, int out_size, void* d_ws, size_t ws_size,
                              hipStream_t stream) {
  const float* x = (const float*)d_in[0];
  const float* W = (const float*)d_in[1];
  float* out = (float*)d_out;

  float* ws = (float*)d_ws;
  const size_t per = (size_t)BATCH * HEADS * SEQ * HD;   // 8388608 floats
  float* wq  = ws;
  float* wk  = ws + per;
  float* wv  = ws + 2 * per;
  float* wg  = ws + 3 * per;                              // 16*64*64
  float* wks = wg + BATCH * HEADS * 64 * 64;              // 16*64

  dim3 g1(NTOT / 128, MTOT / 128);                        // 12 x 128
  qkv_gemm<<<g1, 256, 0, stream>>>(x, W, wq, wk, wv);

  kv_agg<<<BATCH * HEADS, 512, 0, stream>>>(wk, wv, wg, wks);

  dim3 g3(BATCH * HEADS, SEQ / 128);                      // 16 x 64
  out_kernel<<<g3, 256, 0, stream>>>(wq, wg, wks, out);
}